// Biholomorphic_k8_45921790329591
// MI455X (gfx1250) — compile-verified
//
#include <hip/hip_runtime.h>

// ---------------------------------------------------------------------------
// Biholomorphic k=8 feature map, MI455X (gfx1250, wave32).
//
// Stage 1: 256 x 495 complex degree-8 monomials zz  -> workspace (~1 MB).
// Stage 2: Hermitian outer product zz zz^H, packed upper triangle, via
//          V_WMMA_F32_16X16X4_F32 (K=2 used, K=2 padded with zeros):
//            Re tile = [zr_i zi_i] x [zr_j; zi_j]
//            Im tile = [zi_i -zr_i] x [zr_j; zi_j]
//          Output: 251.5 MB of f32 stores -> pure HBM-store bound (~11 us).
// ---------------------------------------------------------------------------

#define NCOORD   5
#define DEG      8
#define NM       495          // C(12,8) monomials
#define STRIDE   512          // padded zz row stride (zero-filled 495..511)
#define NPAIR    123060       // Im offset; Re part holds 495*496/2 = 122760 pairs
#define OUT_PER_B 245625      // NPAIR + (NPAIR - 495)
#define NTILE    31           // ceil(495/16)
#define NTP      496          // 31*32/2 upper-tri tile pairs
#define WAVES_PER_BLOCK 8

typedef __attribute__((ext_vector_type(2))) float v2f;
typedef __attribute__((ext_vector_type(8))) float v8f;

// ---------------------------------------------------------------------------
// Stage 1: per-batch monomials. One block per batch element, 512 threads.
// Thread m unranks the m-th exponent vector (e0..e4), sum 8, in the
// lexicographic order of sorted multi-indices, then computes prod z_c^{e_c}.
// ---------------------------------------------------------------------------
__global__ __launch_bounds__(STRIDE) void bk8_monomials(
    const float* __restrict__ z_re, const float* __restrict__ z_im,
    float* __restrict__ zz_re, float* __restrict__ zz_im) {
  const int b = blockIdx.x;
  const int m = threadIdx.x;
  const size_t base = (size_t)b * STRIDE;
  if (m >= NM) {  // zero-pad so padded matrix row i=495 contributes nothing
    zz_re[base + m] = 0.0f;
    zz_im[base + m] = 0.0f;
    return;
  }
  // Unrank exponent vector: e0 descending, then e1 descending, ... matches
  // lexicographic order of sorted index tuples (0,0,..,0),(0,..,0,1),...
  int e0s = 0, e1s = 0, e2s = 0, e3s = 0, e4s = 0;
  {
    int cnt = 0;
    bool found = false;
    for (int e0 = DEG; e0 >= 0 && !found; --e0)
      for (int e1 = DEG - e0; e1 >= 0 && !found; --e1)
        for (int e2 = DEG - e0 - e1; e2 >= 0 && !found; --e2)
          for (int e3 = DEG - e0 - e1 - e2; e3 >= 0 && !found; --e3) {
            if (cnt == m) {
              e0s = e0; e1s = e1; e2s = e2; e3s = e3;
              e4s = DEG - e0 - e1 - e2 - e3;
              found = true;
            }
            ++cnt;
          }
  }
  int e[NCOORD] = {e0s, e1s, e2s, e3s, e4s};
  float wr = 1.0f, wi = 0.0f;
#pragma unroll
  for (int c = 0; c < NCOORD; ++c) {
    const float zr = z_re[b * NCOORD + c];
    const float zi = z_im[b * NCOORD + c];
    for (int t = 0; t < e[c]; ++t) {
      const float nr = wr * zr - wi * zi;
      const float ni = wr * zi + wi * zr;
      wr = nr; wi = ni;
    }
  }
  zz_re[base + m] = wr;
  zz_im[base + m] = wi;
}

// ---------------------------------------------------------------------------
// Stage 2: one wave32 per (batch, tile-pair). Two f32 WMMAs produce the
// 16x16 Re and Im tiles; stores are b32, row-contiguous per half-wave.
// ---------------------------------------------------------------------------
__global__ __launch_bounds__(32 * WAVES_PER_BLOCK) void bk8_outer_wmma(
    const float* __restrict__ zz_re, const float* __restrict__ zz_im,
    float* __restrict__ out) {
  const int b    = blockIdx.y;
  const int wave = threadIdx.x >> 5;
  const int lane = threadIdx.x & 31;
  const int tp   = blockIdx.x * WAVES_PER_BLOCK + wave;  // 0..NTP-1 exactly

  // Invert tp -> (ti, tj), tj >= ti, over NTILE x NTILE upper triangle.
  int ti = 0, off = 0;
  while (off + (NTILE - ti) <= tp) { off += NTILE - ti; ++ti; }
  const int tj = ti + (tp - off);
  const int i0 = ti << 4, j0 = tj << 4;

  const float* __restrict__ zr = zz_re + (size_t)b * STRIDE;
  const float* __restrict__ zi = zz_im + (size_t)b * STRIDE;

  // A (16x4 f32): lanes 0-15 hold M=lane with K0 in VGPR0, K1 in VGPR1;
  // lanes 16-31 hold K2/K3 -> zero padding. B (4x16): mirrored layout.
  v2f a_re = {0.0f, 0.0f}, a_im = {0.0f, 0.0f}, bmat = {0.0f, 0.0f};
  if (lane < 16) {
    const float ir = zr[i0 + lane], ii = zi[i0 + lane];
    a_re[0] = ir;  a_re[1] = ii;    // Re = zr_i*zr_j + zi_i*zi_j
    a_im[0] = ii;  a_im[1] = -ir;   // Im = zi_i*zr_j - zr_i*zi_j
    bmat[0] = zr[j0 + lane];
    bmat[1] = zi[j0 + lane];
  }

  v8f c0 = {};
  v8f dre = __builtin_amdgcn_wmma_f32_16x16x4_f32(
      false, a_re, false, bmat, (short)0, c0, false, false);
  v8f dim = __builtin_amdgcn_wmma_f32_16x16x4_f32(
      false, a_im, false, bmat, (short)0, c0, false, false);

  // D layout: VGPR v holds row M = v + (lane>=16 ? 8 : 0), col N = lane%16.
  const int n     = lane & 15;
  const int rbase = (lane >> 4) << 3;  // 0 or 8
  const int j     = j0 + n;
  float* __restrict__ ob = out + (size_t)b * OUT_PER_B;

  if (ti < tj && ti < NTILE - 1 && tj < NTILE - 1) {
    // Fast path: full strictly-upper interior tile -> unconditional stores.
#pragma unroll
    for (int v = 0; v < 8; ++v) {
      const int i = i0 + rbase + v;
      const int p = i * NM - (i * (i - 1)) / 2 + (j - i);
      ob[p] = dre[v];
      ob[NPAIR + p - (i + 1)] = dim[v];
    }
  } else {
    // Diagonal / edge tiles: mask i<NM, j<NM, j>=i (and j>i for imag).
#pragma unroll
    for (int v = 0; v < 8; ++v) {
      const int i = i0 + rbase + v;
      if (i < NM && j < NM && j >= i) {
        const int p = i * NM - (i * (i - 1)) / 2 + (j - i);
        ob[p] = dre[v];
        if (j > i) ob[NPAIR + p - (i + 1)] = dim[v];
      }
    }
  }
}

// ---------------------------------------------------------------------------
extern "C" void kernel_launch(void* const* d_in, const int* in_sizes, int n_in,
                              void* d_out, int out_size, void* d_ws,
                              size_t ws_size, hipStream_t stream) {
  const float* z_re = (const float*)d_in[0];
  const float* z_im = (const float*)d_in[1];
  float* out = (float*)d_out;
  const int B = in_sizes[0] / NCOORD;  // 256 per reference

  float* zz_re = (float*)d_ws;
  float* zz_im = zz_re + (size_t)B * STRIDE;

  bk8_monomials<<<B, STRIDE, 0, stream>>>(z_re, z_im, zz_re, zz_im);

  dim3 grid(NTP / WAVES_PER_BLOCK, B);  // 62 x 256, 8 waves/block -> 496 pairs
  bk8_outer_wmma<<<grid, 32 * WAVES_PER_BLOCK, 0, stream>>>(zz_re, zz_im, out);
}